// VisualPromptEncoder_32787780338119
// MI455X (gfx1250) — compile-verified
//
#include <hip/hip_runtime.h>

// VisualPromptEncoder eval-path, reformulated:
//   bilinear-resize + integral-image box means  ==  F[b] (C x 1600) @ W[b]^T (1600 x N)
// with W[n, i*40+j] = Wy[n,i] * Wx[n,j] * (valid ? 1/max(area,1) : 0),
// where Wx/Wy are accumulated 1-D bilinear resize tap weights over the box span.
// The GEMM runs on V_WMMA_F32_16X16X4_F32 (full-f32 matrix core path, wave32).

#define NUM_CLASSES 80
#define B_  2
#define C_  128
#define H_  40
#define W_  40
#define N_  32
#define K_  (H_ * W_)   // 1600
#define RES 640

typedef __attribute__((ext_vector_type(2))) float v2f;
typedef __attribute__((ext_vector_type(8))) float v8f;

// ---------------------------------------------------------------------------
// Kernel 1: per-box separable bilinear-sum weights -> W[b*N+n][1600], valid flag.
// One block (2 waves) per box. Wave 0 accumulates Wx, wave 1 accumulates Wy,
// both via LDS float atomics (ds_add_f32). Then the outer product (scaled by
// 1/area, zeroed when invalid) is written to global scratch.
// ---------------------------------------------------------------------------
__global__ __launch_bounds__(64) void vpe_build_weights(
    const float* __restrict__ boxes,
    float* __restrict__ Wout,      // [B*N, K_]
    float* __restrict__ validOut)  // [B*N]
{
    const int box = blockIdx.x;        // 0 .. B*N-1
    const int tid = threadIdx.x;

    __shared__ float sWx[W_];
    __shared__ float sWy[H_];
    if (tid < W_) { sWx[tid] = 0.0f; sWy[tid] = 0.0f; }
    __syncthreads();

    const float* bb = boxes + (size_t)box * 4;
    // jnp.round == round-half-even == rintf (default RNE), then clip to [0, RES]
    int x1 = (int)rintf(bb[0]); x1 = min(max(x1, 0), RES);
    int y1 = (int)rintf(bb[1]); y1 = min(max(y1, 0), RES);
    int x2 = (int)rintf(bb[2]); x2 = min(max(x2, 0), RES);
    int y2 = (int)rintf(bb[3]); y2 = min(max(y2, 0), RES);
    const bool  valid = (x1 < x2) && (y1 < y2);
    const float area  = (float)((x2 - x1) * (y2 - y1));
    const float scale = valid ? (1.0f / fmaxf(area, 1.0f)) : 0.0f;

    // Output pixel x maps to input coord u = (x+0.5)/16 - 0.5; 2-tap triangle
    // kernel with clamped indices (== JAX's drop+renormalize edge handling).
    if (tid < 32) {                         // wave 0: X axis
        for (int x = x1 + tid; x < x2; x += 32) {
            float u  = ((float)x + 0.5f) * 0.0625f - 0.5f;
            float fl = floorf(u);
            float fr = u - fl;
            int   i0 = (int)fl;
            int   j0 = min(max(i0,     0), W_ - 1);
            int   j1 = min(max(i0 + 1, 0), W_ - 1);
            atomicAdd(&sWx[j0], 1.0f - fr);
            atomicAdd(&sWx[j1], fr);
        }
    } else {                                // wave 1: Y axis
        for (int y = y1 + (tid - 32); y < y2; y += 32) {
            float u  = ((float)y + 0.5f) * 0.0625f - 0.5f;
            float fl = floorf(u);
            float fr = u - fl;
            int   i0 = (int)fl;
            int   j0 = min(max(i0,     0), H_ - 1);
            int   j1 = min(max(i0 + 1, 0), H_ - 1);
            atomicAdd(&sWy[j0], 1.0f - fr);
            atomicAdd(&sWy[j1], fr);
        }
    }
    __syncthreads();

    float* wrow = Wout + (size_t)box * K_;
    for (int e = tid; e < K_; e += 64)
        wrow[e] = sWy[e / W_] * sWx[e % W_] * scale;

    if (tid == 0) validOut[box] = valid ? 1.0f : 0.0f;
}

// ---------------------------------------------------------------------------
// Kernel 2: pooled[b, n, c] = sum_k F[b, c, k] * W[b, n, k]
// One wave32 per 16(ch) x 16(box) tile, K=1600 in steps of 4 on
// V_WMMA_F32_16X16X4_F32.
//
// Fragment layouts (ISA 7.12.2), lane = 16*half + ln:
//   A (16x4 MxK):  a[v]  = A[ln, k4 + 2*half + v]      (v = 0,1)
//   B (4x16 KxN):  bv[v] = B[k4 + 2*half + v, ln] = W[n=ln, k4 + 2*half + v]
//   D (16x16):     acc[r] = D[M = r + 8*half, N = ln]
// Both A and B fragments are 2 consecutive K elements -> single b64 loads.
// ---------------------------------------------------------------------------
__global__ __launch_bounds__(32) void vpe_wmma_pool(
    const float* __restrict__ F,      // [B, C_, K_]
    const float* __restrict__ Wm,     // [B, N_, K_]
    float* __restrict__ pooled)       // [B, N_, C_]
{
    int id    = blockIdx.x;
    int ctile = id & 7;   id >>= 3;   // C_/16 = 8
    int ntile = id & 1;   id >>= 1;   // N_/16 = 2
    int b     = id;

    const int lane = threadIdx.x;
    const int half = lane >> 4;
    const int ln   = lane & 15;

    const float* Arow = F  + ((size_t)b * C_ + ctile * 16 + ln) * K_ + 2 * half;
    const float* Brow = Wm + ((size_t)b * N_ + ntile * 16 + ln) * K_ + 2 * half;

    v8f acc = {};
#pragma unroll 8
    for (int k = 0; k < K_; k += 4) {
        v2f a  = *(const v2f*)(Arow + k);
        v2f bv = *(const v2f*)(Brow + k);
        // 8 args: (neg_a, A, neg_b, B, c_mod, C, reuse_a, reuse_b)
        acc = __builtin_amdgcn_wmma_f32_16x16x4_f32(
            false, a, false, bv, (short)0, acc, false, false);
    }

    // D element (r, half, ln) -> pooled[b][ntile*16 + ln][ctile*16 + 8*half + r]
    float* out = pooled + ((size_t)b * N_ + ntile * 16 + ln) * C_
                        + ctile * 16 + 8 * half;
#pragma unroll
    for (int r = 0; r < 8; ++r) out[r] = acc[r];
}

// ---------------------------------------------------------------------------
// Kernel 3: per-(batch, class) mean over valid boxes of that class.
// Grid = B*80 blocks, 128 threads (one per channel). Writes every output
// element (output buffer is poisoned by the harness).
// ---------------------------------------------------------------------------
__global__ __launch_bounds__(C_) void vpe_class_mean(
    const float* __restrict__ pooled,  // [B, N_, C_]
    const float* __restrict__ valid,   // [B*N_]
    const int*   __restrict__ cls,     // [B*N_]
    float* __restrict__ out)           // [B, 80, C_]
{
    const int k = blockIdx.x % NUM_CLASSES;
    const int b = blockIdx.x / NUM_CLASSES;
    const int c = threadIdx.x;

    float sum = 0.0f, cnt = 0.0f;
    for (int n = 0; n < N_; ++n) {
        const int idx = b * N_ + n;
        if (cls[idx] == k && valid[idx] > 0.5f) {
            sum += pooled[(size_t)idx * C_ + c];
            cnt += 1.0f;
        }
    }
    out[((size_t)b * NUM_CLASSES + k) * C_ + c] = (cnt > 0.0f) ? (sum / cnt) : 0.0f;
}

// ---------------------------------------------------------------------------
extern "C" void kernel_launch(void* const* d_in, const int* in_sizes, int n_in,
                              void* d_out, int out_size, void* d_ws, size_t ws_size,
                              hipStream_t stream) {
    const float* features = (const float*)d_in[0];  // [2,128,40,40] f32
    const float* boxes    = (const float*)d_in[1];  // [2,32,4]      f32
    const int*   gtcls    = (const int*)  d_in[2];  // [2,32]        int
    float*       out      = (float*)d_out;          // [2,80,128]    f32

    // Scratch layout (floats): W [B*N*1600] | valid [B*N] | pooled [B*N*C]
    // = 102400 + 64 + 8192 floats ~= 443 KB.
    float* Wf     = (float*)d_ws;
    float* valid  = Wf + (size_t)B_ * N_ * K_;
    float* pooled = valid + (size_t)B_ * N_;

    vpe_build_weights<<<B_ * N_, 64, 0, stream>>>(boxes, Wf, valid);
    vpe_wmma_pool<<<B_ * (N_ / 16) * (C_ / 16), 32, 0, stream>>>(features, Wf, pooled);
    vpe_class_mean<<<B_ * NUM_CLASSES, C_, 0, stream>>>(pooled, valid, gtcls, out);
}